// SpatialWeightsEsDot_11304353923147
// MI455X (gfx1250) — compile-verified
//
#include <hip/hip_runtime.h>

// ---------------------------------------------------------------------------
// Problem: u[2,64,128,128] f32, w_phi[64,64] f32.
// phi = w_phi x u (1x1 conv)  -> WMMA f32 GEMM (kernel A)
// score(p,o) = [p+o in image] * sum_{d in 3x3} G(p+d,o),
//   G(q,o) = sum_c phi0(q,c)*phi0(q+o,c)   (phi0 = phi zero-extended)
// out = softmax over the 49 offsets, laid out [b,h,w,1,49].
// ---------------------------------------------------------------------------

typedef __attribute__((ext_vector_type(2))) float v2f;
typedef __attribute__((ext_vector_type(8))) float v8f;

#define HW 128
#define NPIX (HW * HW)          // 16384 pixels per batch
#define NC 64                   // channels
#define WOFF 49                 // 7x7 window

// ------------------------------ Kernel A -----------------------------------
// phi_t[b][y][x][c] = sum_i w[c,i] * u[b,i,y,x]
// One wave handles 16 pixels x all 64 out-channels via V_WMMA_F32_16X16X4_F32.
// A tile: 16 pixels (M) x 4 in-ch (K); B tile: 4 in-ch (K) x 16 out-ch (N).
__global__ __launch_bounds__(256) void phi_wmma_kernel(
    const float* __restrict__ u, const float* __restrict__ w,
    float* __restrict__ phi) {
  const int lane = threadIdx.x & 31;
  const int wave = threadIdx.x >> 5;
  const int group = blockIdx.x * 8 + wave;      // 2048 groups of 16 pixels
  const int pix0 = group * 16;                  // flat over b*h*w (batch-aligned)
  const int b = pix0 >> 14;
  const int sp0 = pix0 & (NPIX - 1);
  const float* ub = u + (size_t)b * NC * NPIX;

  const int m  = lane & 15;                     // M row (pixel) / N col (chan)
  const int hl = lane >> 4;                     // half-wave selects K pair
  const int kh = hl * 2;

  // Preload full A strip: 16 pixels x 64 in-channels (32 VGPRs).
  // A layout: vgpr v, lanes0-15: K=v ; lanes16-31: K=2+v ; M=lane&15.
  float a[16][2];
#pragma unroll
  for (int ks = 0; ks < 16; ++ks) {
    const int k0 = ks * 4;
    a[ks][0] = ub[(size_t)(k0 + kh + 0) * NPIX + sp0 + m];
    a[ks][1] = ub[(size_t)(k0 + kh + 1) * NPIX + sp0 + m];
  }

#pragma unroll
  for (int nt = 0; nt < 4; ++nt) {              // out-channel tiles
    const int co = nt * 16;
    v8f acc = {};
#pragma unroll
    for (int ks = 0; ks < 16; ++ks) {
      const int k0 = ks * 4;
      // B layout (mirror of A): vgpr v, lanes0-15: K=v ; lanes16-31: K=2+v.
      v2f av = {a[ks][0], a[ks][1]};
      v2f bv = {w[(co + m) * NC + k0 + kh + 0],
                w[(co + m) * NC + k0 + kh + 1]};
      acc = __builtin_amdgcn_wmma_f32_16x16x4_f32(
          /*neg_a=*/false, av, /*neg_b=*/false, bv,
          /*c_mod=*/(short)0, acc, /*reuse_a=*/false, /*reuse_b=*/false);
    }
    // D layout: vgpr v, lanes0-15: M=v ; lanes16-31: M=v+8 ; N=lane&15.
#pragma unroll
    for (int v = 0; v < 8; ++v) {
      const int mrow = v + 8 * hl;
      phi[(size_t)(pix0 + mrow) * NC + co + m] = acc[v];
    }
  }
}

// ------------------------------ Kernel B -----------------------------------
// 16x16 spatial tile per block (256 threads). Two 32-channel chunks of the
// phi tile (+halo 4) staged in LDS; G(q,o) accumulated in LDS for tile+halo1;
// then per-pixel 3x3 box sum + boundary mask + softmax.
#define TILE 16
#define PR (TILE + 8)           // 24: phi region side (halo 4)
#define GR (TILE + 2)           // 18: G region side (halo 1)
#define CCH 32                  // channel chunk

__global__ __launch_bounds__(256) void scores_softmax_kernel(
    const float* __restrict__ phi, float* __restrict__ out) {
  __shared__ __align__(16) float sphi[PR * PR * CCH];   // 73,728 B
  __shared__ __align__(16) float sG[GR * GR * WOFF];    // 63,504 B

  const int tx0 = blockIdx.x * TILE;
  const int ty0 = blockIdx.y * TILE;
  const int b   = blockIdx.z;
  const float* phb = phi + (size_t)b * NPIX * NC;
  const int tid = threadIdx.x;

  for (int ck = 0; ck < 2; ++ck) {
    // ---- stage phi chunk (24x24 pixels x 32 channels), zero outside image
    for (int idx = tid; idx < PR * PR * (CCH / 4); idx += 256) {
      const int pix = idx >> 3;            // 0..575
      const int c4  = (idx & 7) * 4;       // float4 chunk within 32 ch
      const int ry = pix / PR, rx = pix - ry * PR;
      const int gy = ty0 - 4 + ry, gx = tx0 - 4 + rx;
      float4 v = make_float4(0.f, 0.f, 0.f, 0.f);
      if ((unsigned)gy < (unsigned)HW && (unsigned)gx < (unsigned)HW)
        v = *(const float4*)(phb + ((size_t)(gy * HW + gx)) * NC + ck * CCH + c4);
      *(float4*)(&sphi[pix * CCH + c4]) = v;
    }
    __syncthreads();

    // ---- accumulate G(q,o) over this channel chunk
    for (int t = tid; t < GR * GR * WOFF; t += 256) {
      const int q = t / WOFF, o = t - q * WOFF;
      const int qy = q / GR, qx = q - qy * GR;
      const int oy = o / 7 - 3, ox = o % 7 - 3;
      const int l0 = ((qy + 3) * PR + (qx + 3)) * CCH;
      const int l1 = ((qy + 3 + oy) * PR + (qx + 3 + ox)) * CCH;
      float s = 0.f;
#pragma unroll
      for (int c = 0; c < CCH; c += 4) {
        const float4 pa = *(const float4*)(&sphi[l0 + c]);
        const float4 pb = *(const float4*)(&sphi[l1 + c]);
        s += pa.x * pb.x + pa.y * pb.y + pa.z * pb.z + pa.w * pb.w;
      }
      if (ck == 0) sG[t] = s; else sG[t] += s;
    }
    __syncthreads();
  }

  // ---- per-pixel: 3x3 box sum of G, boundary mask, softmax over 49
  const int ty = tid >> 4, tx = tid & 15;
  const int y = ty0 + ty, x = tx0 + tx;

  float sc[WOFF];
#pragma unroll
  for (int o = 0; o < WOFF; ++o) {
    const int oy = o / 7 - 3, ox = o % 7 - 3;
    float s = 0.f;
    if ((unsigned)(y + oy) < (unsigned)HW && (unsigned)(x + ox) < (unsigned)HW) {
#pragma unroll
      for (int dy = 0; dy < 3; ++dy)
#pragma unroll
        for (int dx = 0; dx < 3; ++dx)
          s += sG[(((ty + dy) * GR) + (tx + dx)) * WOFF + o];
    }
    sc[o] = s;   // out-of-window offsets contribute exact 0 (softmax sees 0)
  }

  float mx = sc[0];
#pragma unroll
  for (int o = 1; o < WOFF; ++o) mx = fmaxf(mx, sc[o]);
  float sum = 0.f;
#pragma unroll
  for (int o = 0; o < WOFF; ++o) { const float e = __expf(sc[o] - mx); sc[o] = e; sum += e; }
  const float inv = 1.f / sum;

  float* op = out + ((size_t)((b * HW + y) * HW + x)) * WOFF;
#pragma unroll
  for (int o = 0; o < WOFF; ++o) op[o] = sc[o] * inv;
}

// ------------------------------ launcher -----------------------------------
extern "C" void kernel_launch(void* const* d_in, const int* in_sizes, int n_in,
                              void* d_out, int out_size, void* d_ws, size_t ws_size,
                              hipStream_t stream) {
  const float* u = (const float*)d_in[0];   // [2,64,128,128]
  const float* w = (const float*)d_in[1];   // [64,64]
  float* out = (float*)d_out;               // [2,128,128,1,49]
  float* phi = (float*)d_ws;                // 2*16384*64 f32 = 8 MB scratch

  // Kernel A: 2048 waves (one per 16-pixel group), 8 waves per block.
  phi_wmma_kernel<<<256, 256, 0, stream>>>(u, w, phi);

  // Kernel B: one block per 16x16 spatial tile per batch.
  dim3 grid(HW / TILE, HW / TILE, 2);
  scores_softmax_kernel<<<grid, 256, 0, stream>>>(phi, out);
}